// GCN_59794534695170
// MI455X (gfx1250) — compile-verified
//
#include <hip/hip_runtime.h>

typedef float v2f __attribute__((ext_vector_type(2)));
typedef float v8f __attribute__((ext_vector_type(8)));

// ---------------------------------------------------------------------------
// Degree accumulation: one thread per edge, hardware f32 atomics.
// ---------------------------------------------------------------------------
__global__ void k_degrees(const int* __restrict__ src, const int* __restrict__ dst,
                          float* __restrict__ deg_out, float* __restrict__ deg_in, int E) {
    int e = blockIdx.x * blockDim.x + threadIdx.x;
    if (e < E) {
        unsafeAtomicAdd(&deg_out[src[e]], 1.0f);
        unsafeAtomicAdd(&deg_in[dst[e]], 1.0f);
    }
}

// deg -> rsqrt(max(deg,1)) in place for both degree arrays
__global__ void k_rsqrt_deg(float* __restrict__ a, float* __restrict__ b, int n) {
    int i = blockIdx.x * blockDim.x + threadIdx.x;
    if (i < n) {
        a[i] = rsqrtf(fmaxf(a[i], 1.0f));
        b[i] = rsqrtf(fmaxf(b[i], 1.0f));
    }
}

// ---------------------------------------------------------------------------
// Edge scatter, 128 features: one wave32 per edge, one float4 per lane.
// agg[dst] += x[src] * rsqrt(deg_out[src])
// ---------------------------------------------------------------------------
__global__ void k_scatter128(const float4* __restrict__ xv,
                             const int* __restrict__ src, const int* __restrict__ dst,
                             const float* __restrict__ deg_out_inv,
                             float* __restrict__ agg, int E) {
    int gid  = blockIdx.x * blockDim.x + threadIdx.x;
    int e    = gid >> 5;
    int lane = gid & 31;
    if (e >= E) return;
    int   s  = src[e];
    int   d  = dst[e];
    float sc = deg_out_inv[s];
    float4 v = xv[(size_t)s * 32 + lane];          // coalesced global_load_b128
    float* o = agg + (size_t)d * 128 + lane * 4;
    unsafeAtomicAdd(o + 0, v.x * sc);
    unsafeAtomicAdd(o + 1, v.y * sc);
    unsafeAtomicAdd(o + 2, v.z * sc);
    unsafeAtomicAdd(o + 3, v.w * sc);
}

// Edge scatter, 64 features: one wave32 per edge, one float2 per lane.
__global__ void k_scatter64(const float2* __restrict__ hv,
                            const int* __restrict__ src, const int* __restrict__ dst,
                            const float* __restrict__ deg_out_inv,
                            float* __restrict__ agg, int E) {
    int gid  = blockIdx.x * blockDim.x + threadIdx.x;
    int e    = gid >> 5;
    int lane = gid & 31;
    if (e >= E) return;
    int   s  = src[e];
    int   d  = dst[e];
    float sc = deg_out_inv[s];
    float2 v = hv[(size_t)s * 32 + lane];
    float* o = agg + (size_t)d * 64 + lane * 2;
    unsafeAtomicAdd(o + 0, v.x * sc);
    unsafeAtomicAdd(o + 1, v.y * sc);
}

// ---------------------------------------------------------------------------
// WMMA GEMM: out[N,64] = op( (A * rsqrt(deg_in)[:,None]) @ W + bias )
// A is [n_rows, K] (K = 128 or 64), W is [K, 64] staged in LDS.
// Block = 256 threads = 8 waves; each wave owns one 16x16 tile.
// Uses V_WMMA_F32_16X16X4_F32 (full fp32 matrix pipe, ISA 7.12.2 layouts):
//   A 16x4 : lanes 0-15 hold K=k0,k0+1 ; lanes 16-31 hold K=k0+2,k0+3
//   B 4x16 : VGPR v, lane-half h -> row K=k0+2h+v, col = lane&15
//   C 16x16: VGPR i, lane-half h -> row i+8h, col = lane&15
// ---------------------------------------------------------------------------
template <int K, bool RELU>
__global__ void __launch_bounds__(256)
k_gemm_wmma(const float* __restrict__ A, const float* __restrict__ W,
            const float* __restrict__ bias, const float* __restrict__ din_inv,
            float* __restrict__ out, int n_rows) {
    __shared__ float Ws[K * 64];

    // cooperative LDS fill of W (K*64 floats, float4 vectorized)
    {
        const float4* Wv  = (const float4*)W;
        float4*       Wsv = (float4*)Ws;
        for (int i = threadIdx.x; i < K * 16; i += 256) Wsv[i] = Wv[i];
    }
    __syncthreads();

    int lane  = threadIdx.x & 31;
    int wv    = threadIdx.x >> 5;
    int rowTileBase = blockIdx.x * 32 + (wv >> 2) * 16;  // 2 row tiles per block
    int colBase     = (wv & 3) * 16;                     // 4 col tiles per block
    int l16   = lane & 15;
    int khalf = lane >> 4;                               // 0 or 1

    int mrow   = rowTileBase + l16;
    int mclamp = mrow < n_rows ? mrow : n_rows - 1;      // branchless, EXEC stays full
    float din  = din_inv[mclamp];
    int  bcol  = colBase + l16;

    const float* Arow = A + (size_t)mclamp * K;
    v8f c = {};
#pragma unroll
    for (int k0 = 0; k0 < K; k0 += 4) {
        int ka = k0 + khalf * 2;
        v2f a;
        a.x = Arow[ka]     * din;                        // fuse dest-norm scaling
        a.y = Arow[ka + 1] * din;
        v2f b;
        b.x = Ws[ka * 64 + bcol];                        // ds_load from staged W
        b.y = Ws[(ka + 1) * 64 + bcol];
        c = __builtin_amdgcn_wmma_f32_16x16x4_f32(
                /*neg_a=*/false, a, /*neg_b=*/false, b,
                /*c_mod=*/(short)0, c, /*reuse_a=*/false, /*reuse_b=*/false);
    }

    float bz = bias[bcol];
#pragma unroll
    for (int i = 0; i < 8; i++) {
        int r = rowTileBase + i + khalf * 8;
        if (r < n_rows) {
            float v = c[i] + bz;
            if (RELU) v = fmaxf(v, 0.0f);
            out[(size_t)r * 64 + bcol] = v;
        }
    }
}

// ---------------------------------------------------------------------------
// Orchestration. Workspace layout (floats):
//   [0, N)            rsqrt(deg_out)
//   [N, 2N)           rsqrt(deg_in)
//   [2N, 2N+128N)     agg1  (layer-1 accumulator; first 64N reused as agg2)
//   [2N+128N, +64N)   h1    (layer-1 output)
// ---------------------------------------------------------------------------
extern "C" void kernel_launch(void* const* d_in, const int* in_sizes, int n_in,
                              void* d_out, int out_size, void* d_ws, size_t ws_size,
                              hipStream_t stream) {
    const float* x   = (const float*)d_in[0];
    const int*   src = (const int*)d_in[1];
    const int*   dst = (const int*)d_in[2];
    const float* W1  = (const float*)d_in[3];
    const float* b1  = (const float*)d_in[4];
    const float* W2  = (const float*)d_in[5];
    const float* b2  = (const float*)d_in[6];
    float*       out = (float*)d_out;

    const int N = in_sizes[0] / 128;
    const int E = in_sizes[1];

    float* ws          = (float*)d_ws;
    float* deg_out_inv = ws;
    float* deg_in_inv  = ws + N;
    float* agg1        = ws + 2 * (size_t)N;            // N*128 floats (agg2 aliases)
    float* h1          = agg1 + (size_t)N * 128;        // N*64 floats

    // zero degree accumulators + agg1 in one stream-ordered memset
    hipMemsetAsync(ws, 0, (size_t)(2 + 128) * N * sizeof(float), stream);

    k_degrees<<<(E + 255) / 256, 256, 0, stream>>>(src, dst, deg_out_inv, deg_in_inv, E);
    k_rsqrt_deg<<<(N + 255) / 256, 256, 0, stream>>>(deg_out_inv, deg_in_inv, N);

    // layer 1: scatter-aggregate then WMMA GEMM (+bias, ReLU)
    {
        long long T = (long long)E * 32;
        k_scatter128<<<(int)((T + 255) / 256), 256, 0, stream>>>(
            (const float4*)x, src, dst, deg_out_inv, agg1, E);
    }
    int gemmBlocks = (N + 31) / 32;
    k_gemm_wmma<128, true><<<gemmBlocks, 256, 0, stream>>>(agg1, W1, b1, deg_in_inv, h1, N);

    // layer 2: re-zero aliased accumulator, scatter, WMMA GEMM (+bias)
    hipMemsetAsync(agg1, 0, (size_t)N * 64 * sizeof(float), stream);
    {
        long long T = (long long)E * 32;
        k_scatter64<<<(int)((T + 255) / 256), 256, 0, stream>>>(
            (const float2*)h1, src, dst, deg_out_inv, agg1, E);
    }
    k_gemm_wmma<64, false><<<gemmBlocks, 256, 0, stream>>>(agg1, W2, b2, deg_in_inv, out, N);
}